// StoppingTimeProximityLoss_81320910782865
// MI455X (gfx1250) — compile-verified
//
#include <hip/hip_runtime.h>

#define N_SAMPLES 2048
#define T_LEN     1024
#define C_CLS     24
#define MAX_PAIRS 256

typedef __attribute__((ext_vector_type(2))) float v2f;
typedef __attribute__((ext_vector_type(8))) float v8f;

// -------- Workspace layout (float units) --------
// [0    .. 1023]  per-block classification partials (1024 blocks)
// [1024 .. 2047]  per-block earliness partials
// [2048 .. 2079]  per-class proximity partials (24 used)
// [2080 .. 4127]  t_final per n (int)
// [4128 .. 6175]  v per n (float)
// [6176 .. 8223]  cls per n (int)
#define WS_CL    0
#define WS_ER    1024
#define WS_PROX  2048
#define WS_TF    2080
#define WS_V     4128
#define WS_CLS   6176

// ---------------------------------------------------------------------------
// Kernel 1: streaming gather + WMMA-pipe reduction of classification &
// earliness terms. 1024 blocks x 256 threads; each thread handles exactly
// 2 elements per iteration, 4 iterations -> exact coverage of N*T = 2^21.
// EXEC is all-ones everywhere (required for WMMA).
// ---------------------------------------------------------------------------
__global__ __launch_bounds__(256) void k_main_reduce(
    const float* __restrict__ logp,
    const float* __restrict__ tl,
    const int*   __restrict__ yt,
    float* __restrict__ cl_part,
    float* __restrict__ er_part)
{
    const int tid  = threadIdx.x;
    const int gtid = blockIdx.x * 256 + tid;
    const int elemsPerIter = 1024 * 256 * 2;   // 524288
    const float invT = 1.0f / (float)T_LEN;

    v2f ones; ones[0] = 1.0f; ones[1] = 1.0f;  // B = all-ones 4x16
    v8f c_cl = {};                              // f32 accumulators on WMMA pipe
    v8f c_er = {};

    #pragma unroll
    for (int it = 0; it < 4; ++it) {
        const int e = it * elemsPerIter + gtid * 2;   // even, same row n for e,e+1

        const int2   yy = *reinterpret_cast<const int2*>(yt + e);
        const float2 tt = *reinterpret_cast<const float2*>(tl + e);

        const float lp0 = logp[(size_t)e       * C_CLS + yy.x];
        const float lp1 = logp[(size_t)(e + 1) * C_CLS + yy.y];

        const int t0 = e & (T_LEN - 1);
        const int t1 = t0 + 1;

        const float er0 = __expf(lp0) * (1.0f - (float)t0 * invT) * (1.0f - tt.x * invT);
        const float er1 = __expf(lp1) * (1.0f - (float)t1 * invT) * (1.0f - tt.y * invT);

        v2f a_cl; a_cl[0] = -lp0; a_cl[1] = -lp1;
        v2f a_er; a_er[0] =  er0; a_er[1] =  er1;

        // D[m,n] = sum_k A[m,k] + C[m,n]  (B == ones): grand-total accumulator.
        c_cl = __builtin_amdgcn_wmma_f32_16x16x4_f32(
                   false, a_cl, false, ones, (short)0, c_cl, false, false);
        c_er = __builtin_amdgcn_wmma_f32_16x16x4_f32(
                   false, a_er, false, ones, (short)0, c_er, false, false);
    }

    // Column-0 extraction: lanes 0-15 hold sum over rows 0..7 (identical in all
    // 16 lanes), lanes 16-31 hold sum over rows 8..15. xor-16 gives wave total.
    float s_cl = c_cl[0] + c_cl[1] + c_cl[2] + c_cl[3] +
                 c_cl[4] + c_cl[5] + c_cl[6] + c_cl[7];
    float s_er = c_er[0] + c_er[1] + c_er[2] + c_er[3] +
                 c_er[4] + c_er[5] + c_er[6] + c_er[7];
    s_cl += __shfl_xor(s_cl, 16, 32);
    s_er += __shfl_xor(s_er, 16, 32);

    __shared__ float sh_cl[8];
    __shared__ float sh_er[8];
    const int wid = tid >> 5;
    if ((tid & 31) == 0) { sh_cl[wid] = s_cl; sh_er[wid] = s_er; }
    __syncthreads();
    if (tid == 0) {
        float a = 0.0f, b = 0.0f;
        #pragma unroll
        for (int w = 0; w < 8; ++w) { a += sh_cl[w]; b += sh_er[w]; }
        cl_part[blockIdx.x] = a;     // deterministic: fixed tree, no atomics
        er_part[blockIdx.x] = b;
    }
}

// ---------------------------------------------------------------------------
// Kernel 2: per-sample t_final = argmax(timestamps_left <= 0) (first index,
// 0 if none), v = exp(logp[n, t_final, cls]), cls = y_true[n, 0].
// One block per n.
// ---------------------------------------------------------------------------
__global__ __launch_bounds__(256) void k_tfinal(
    const float* __restrict__ tl,
    const int*   __restrict__ yt,
    const float* __restrict__ logp,
    int*   __restrict__ tf_out,
    float* __restrict__ v_out,
    int*   __restrict__ cls_out)
{
    const int n   = blockIdx.x;
    const int tid = threadIdx.x;
    __shared__ int sh[256];

    int best = T_LEN;
    for (int t = tid; t < T_LEN; t += 256) {
        if (tl[(size_t)n * T_LEN + t] <= 0.0f) best = min(best, t);
    }
    sh[tid] = best;
    __syncthreads();
    for (int s = 128; s > 0; s >>= 1) {
        if (tid < s) sh[tid] = min(sh[tid], sh[tid + s]);
        __syncthreads();
    }
    if (tid == 0) {
        int tf = sh[0];
        if (tf == T_LEN) tf = 0;                 // argmax of all-false -> 0
        const int cls = yt[(size_t)n * T_LEN];
        const float v = __expf(logp[((size_t)n * T_LEN + tf) * C_CLS + cls]);
        tf_out[n]  = tf;
        v_out[n]   = v;
        cls_out[n] = cls;
    }
}

// ---------------------------------------------------------------------------
// Kernel 3: proximity reward. combo_idx is the lexicographic index of rank
// pair (a<b) within a class, so "combo_idx < MAX_PAIRS" selects the first
// min(256, size*(size-1)/2) pairs. One block per class.
// ---------------------------------------------------------------------------
__global__ __launch_bounds__(256) void k_pairs(
    const int*   __restrict__ cls,
    const int*   __restrict__ tf,
    const float* __restrict__ v,
    float* __restrict__ prox_part)
{
    const int c   = blockIdx.x;
    const int tid = threadIdx.x;

    __shared__ int members[N_SAMPLES];
    __shared__ int s_cnt;
    if (tid == 0) {                      // rank order == n order (cumsum onehot)
        int cnt = 0;
        for (int n = 0; n < N_SAMPLES; ++n)
            if (cls[n] == c) members[cnt++] = n;
        s_cnt = cnt;
    }
    __syncthreads();

    const int size = s_cnt;
    const long npairs = (long)size * (size - 1) / 2;
    const int  limit  = (int)(npairs < (long)MAX_PAIRS ? npairs : (long)MAX_PAIRS);

    float contrib = 0.0f;
    if (tid < limit) {
        // decode lexicographic pair index -> (a, b)
        int rem = tid;
        int a = 0;
        int rowlen = size - 1;
        while (rem >= rowlen) { rem -= rowlen; --rowlen; ++a; }
        const int b = a + 1 + rem;
        const int i = members[a];
        const int j = members[b];
        const float dtf = (float)(tf[i] - tf[j]) * (1.0f / (float)T_LEN);
        contrib = v[i] * v[j] * dtf * dtf;
    }

    __shared__ float red[256];
    red[tid] = contrib;
    __syncthreads();
    for (int s = 128; s > 0; s >>= 1) {
        if (tid < s) red[tid] += red[tid + s];
        __syncthreads();
    }
    if (tid == 0) prox_part[c] = red[0];
}

// ---------------------------------------------------------------------------
// Kernel 4: deterministic finalize.
// loss = (cl_sum/N - er_sum/N - prox_sum) / 3
// ---------------------------------------------------------------------------
__global__ __launch_bounds__(256) void k_final(
    const float* __restrict__ cl_part,
    const float* __restrict__ er_part,
    const float* __restrict__ prox_part,
    float* __restrict__ out)
{
    const int tid = threadIdx.x;
    __shared__ float r1[256], r2[256], r3[256];

    float a = 0.0f, b = 0.0f;
    for (int i = tid; i < 1024; i += 256) { a += cl_part[i]; b += er_part[i]; }
    float p = (tid < C_CLS) ? prox_part[tid] : 0.0f;

    r1[tid] = a; r2[tid] = b; r3[tid] = p;
    __syncthreads();
    for (int s = 128; s > 0; s >>= 1) {
        if (tid < s) { r1[tid] += r1[tid + s]; r2[tid] += r2[tid + s]; r3[tid] += r3[tid + s]; }
        __syncthreads();
    }
    if (tid == 0) {
        const float cl   = r1[0] * (1.0f / (float)N_SAMPLES);
        const float er   = r2[0] * (1.0f / (float)N_SAMPLES);
        const float prox = r3[0];
        out[0] = (cl - er - prox) * (1.0f / 3.0f);
    }
}

// ---------------------------------------------------------------------------
extern "C" void kernel_launch(void* const* d_in, const int* in_sizes, int n_in,
                              void* d_out, int out_size, void* d_ws, size_t ws_size,
                              hipStream_t stream) {
    const float* logp = (const float*)d_in[0];   // (N, T, C) f32
    const float* tl   = (const float*)d_in[1];   // (N, T)    f32
    const int*   yt   = (const int*)  d_in[2];   // (N, T)    i32
    float* out = (float*)d_out;

    float* w        = (float*)d_ws;
    float* cl_part  = w + WS_CL;
    float* er_part  = w + WS_ER;
    float* prox     = w + WS_PROX;
    int*   tf       = (int*)  (w + WS_TF);
    float* vv       = w + WS_V;
    int*   cls      = (int*)  (w + WS_CLS);

    k_main_reduce<<<1024, 256, 0, stream>>>(logp, tl, yt, cl_part, er_part);
    k_tfinal     <<<N_SAMPLES, 256, 0, stream>>>(tl, yt, logp, tf, vv, cls);
    k_pairs      <<<C_CLS, 256, 0, stream>>>(cls, tf, vv, prox);
    k_final      <<<1, 256, 0, stream>>>(cl_part, er_part, prox, out);
}